// eq_network_82162724372859
// MI455X (gfx1250) — compile-verified
//
#include <hip/hip_runtime.h>

// Equivariant GNN block, algebraically collapsed:
//   seg(small ⊗ t1)[src] · t2  ==  seg(small)[src] * (t1·t2)
// => per-node state shrinks from 4x368 to 16 floats; six weight dots computed
//    once on-device with V_WMMA_F32_16X16X4_F32.

#define NNODE 2048     // NB*NA = 4*512
#define EFIX  40960

typedef __attribute__((ext_vector_type(2))) float v2f;
typedef __attribute__((ext_vector_type(8))) float v8f;

__device__ __constant__ const float SQRT3f     = 1.7320508075688772f;
__device__ __constant__ const float INV_SQRT2f = 0.70710678118654752f;
__device__ __constant__ const float INV_SQRT3f = 0.57735026918962576f;
__device__ __constant__ const float INV_SQRT6f = 0.40824829046386302f;
__device__ __constant__ const float PW1_000f   = 0.70710678118654752f; // sqrt(0.5)
__device__ __constant__ const float PW1_110f   = 0.70710678118654752f;
__device__ __constant__ const float PW1_011f   = 1.2247448713915890f;  // sqrt(1.5)
__device__ __constant__ const float PW1_101f   = 1.2247448713915890f;
__device__ __constant__ const float PW1_111f   = 1.7320508075688772f;  // sqrt(3)
__device__ __constant__ const float PW1_112f   = 2.2360679774997896f;  // sqrt(5)
__device__ __constant__ const float PW2f       = 0.15309310892394863f; // sqrt(3/128)
__device__ __constant__ const float CNf        = 0.31622776601683794f; // 1/sqrt(10) (C112 norm)
__device__ __constant__ const float S3f        = 0.57735026918962576f; // 1/sqrt(3)

// workspace layout (float offsets)
#define DOTS_OFF 0
#define ES_OFF   8
#define HS_OFF   (ES_OFF + NNODE*4)
#define GS_OFF   (HS_OFF + NNODE*4)

__device__ __forceinline__ void atomAdd(float* p, float v) {
    __hip_atomic_fetch_add(p, v, __ATOMIC_RELAXED, __HIP_MEMORY_SCOPE_AGENT);
}

// ---------- zero ES/HS/GS (contiguous, 24*NNODE floats) + out (3*NNODE) ----------
__global__ void k_zero(float* __restrict__ es, float* __restrict__ out) {
    int i = blockIdx.x * blockDim.x + threadIdx.x;
    const int nws = NNODE * 24;
    if (i < nws)                  es[i] = 0.0f;
    else if (i < nws + NNODE * 3) out[i - nws] = 0.0f;
}

// ---------- six weight-weight dot products via V_WMMA_F32_16X16X4_F32 ----------
// A[m,k] = x[4m+k], B[k,n] = y[4n+k]  =>  trace(D) = dot(x, y) (zero padded)
__global__ __launch_bounds__(32) void k_dots(
    const float* __restrict__ t1_000, const float* __restrict__ t1_011,
    const float* __restrict__ t1_101, const float* __restrict__ t1_110,
    const float* __restrict__ t1_111, const float* __restrict__ t1_112,
    const float* __restrict__ t2_0e,  const float* __restrict__ t2_1e,
    const float* __restrict__ t2_1o,  const float* __restrict__ t2_2e,
    float* __restrict__ dots)
{
    const int lane = threadIdx.x;
    const float* xs[6]  = { t1_000, t1_110, t1_011, t1_101, t1_111, t1_112 };
    const float* ys[6]  = { t2_0e,  t2_0e,  t2_1e,  t2_1e,  t2_1o,  t2_2e  };
    const int    lens[6] = { 64, 64, 24, 24, 24, 16 };
    __shared__ float buf[32];

    // 32-bit A 16x4 layout: lanes 0-15 -> K={0,1}, lanes 16-31 -> K={2,3}
    const int m  = (lane < 16) ? lane : (lane - 16);
    const int k0 = (lane < 16) ? 0 : 2;
    const int i0 = 4 * m + k0;

    #pragma unroll
    for (int p = 0; p < 6; ++p) {
        const float* x = xs[p];
        const float* y = ys[p];
        const int L = lens[p];
        v2f a, b;
        a.x = (i0     < L) ? x[i0]     : 0.0f;
        a.y = (i0 + 1 < L) ? x[i0 + 1] : 0.0f;
        b.x = (i0     < L) ? y[i0]     : 0.0f;
        b.y = (i0 + 1 < L) ? y[i0 + 1] : 0.0f;
        v8f c = {};
        c = __builtin_amdgcn_wmma_f32_16x16x4_f32(
                false, a, false, b, (short)0, c, false, false);
        // diagonal of 16x16 f32 D: (m,m) -> m<8: lane m, vgpr m ; m>=8: lane m+16, vgpr m-8
        float v = 0.0f;
        const int dr = (lane < 8) ? lane : ((lane >= 24) ? (lane - 24) : -1);
        #pragma unroll
        for (int r = 0; r < 8; ++r)
            if (dr == r) v = c[r];
        buf[lane] = v;
        __syncthreads();
        if (lane == 0) {
            float s = 0.0f;
            #pragma unroll
            for (int t = 0; t < 32; ++t) s += buf[t];
            dots[p] = s;
        }
        __syncthreads();
    }
}

// ---------- phase 1: ES[d] += INV_SQRT2 * sh(evec)  (seg(esh)) ----------
__global__ void k_edge_sh(const float* __restrict__ r, const int* __restrict__ esrc,
                          const int* __restrict__ edst, float* __restrict__ ES)
{
    int e = blockIdx.x * blockDim.x + threadIdx.x;
    if (e >= EFIX) return;
    int d = edst[e];
    if (d >= NNODE) return;   // padded edge -> dropped segment
    int s = esrc[e];
    float ex = r[3*s+0] - r[3*d+0];
    float ey = r[3*s+1] - r[3*d+1];
    float ez = r[3*s+2] - r[3*d+2];
    atomAdd(&ES[4*d+0], INV_SQRT2f);
    atomAdd(&ES[4*d+1], INV_SQRT2f * SQRT3f * ex);
    atomAdd(&ES[4*d+2], INV_SQRT2f * SQRT3f * ey);
    atomAdd(&ES[4*d+3], INV_SQRT2f * SQRT3f * ez);
}

// ---------- phase 2: HS[n] = sum_c h[n,c,:]  (c=0: ES ; c=1..3: sh(emb[nf])) ----------
__global__ void k_build_hs(const int* __restrict__ nf, const float* __restrict__ emb,
                           const float* __restrict__ ES, float* __restrict__ HS)
{
    int n = blockIdx.x * blockDim.x + threadIdx.x;
    if (n >= NNODE) return;
    int i0 = nf[3*n+0], i1 = nf[3*n+1], i2 = nf[3*n+2];
    HS[4*n+0] = ES[4*n+0] + 3.0f;  // three leading 1's from _sh over the 3 channels
    #pragma unroll
    for (int j = 0; j < 3; ++j)
        HS[4*n+1+j] = ES[4*n+1+j] + SQRT3f * (emb[3*i0+j] + emb[3*i1+j] + emb[3*i2+j]);
}

// ---------- phase 3: GS[d,0:16] += INV_SQRT2 * (c-summed pre-weight edge features) ----------
__global__ void k_edge_mid(const float* __restrict__ r, const int* __restrict__ esrc,
                           const int* __restrict__ edst, const float* __restrict__ HS,
                           float* __restrict__ GS)
{
    int e = blockIdx.x * blockDim.x + threadIdx.x;
    if (e >= EFIX) return;
    int d = edst[e];
    if (d >= NNODE) return;
    int s = esrc[e];
    float s1x = SQRT3f * (r[3*s+0] - r[3*d+0]);
    float s1y = SQRT3f * (r[3*s+1] - r[3*d+1]);
    float s1z = SQRT3f * (r[3*s+2] - r[3*d+2]);

    float4 h = *(const float4*)(HS + 4*s);   // c-summed (a0, a1)
    float A0 = h.x, Ax = h.y, Ay = h.z, Az = h.w;

    float v[16];
    v[0] = PW1_000f * A0;                                   // scalar path (s0 == 1)
    float dt = Ax*s1x + Ay*s1y + Az*s1z;
    v[1] = PW1_110f * INV_SQRT3f * dt;
    float kp = PW1_011f * INV_SQRT3f * A0;                  // p011 = a0 * s1
    v[2] = kp * s1x;  v[3] = kp * s1y;  v[4] = kp * s1z;
    float kq = PW1_101f * INV_SQRT3f;                       // p101 = a1 * s0
    v[5] = kq * Ax;   v[6] = kq * Ay;   v[7] = kq * Az;
    float cx = Ay*s1z - Az*s1y, cy = Az*s1x - Ax*s1z, cz = Ax*s1y - Ay*s1x; // a1 x s1
    float kc = PW1_111f * INV_SQRT6f;
    v[8] = kc * cx;   v[9] = kc * cy;   v[10] = kc * cz;
    float kw = PW1_112f * CNf;                              // C112 contraction (l=2)
    v[11] = kw * (Ax*s1y + Ay*s1x);
    v[12] = kw * (Ay*s1z + Az*s1y);
    v[13] = kw * S3f * (2.0f*Az*s1z - Ax*s1x - Ay*s1y);
    v[14] = kw * (Ax*s1z + Az*s1x);
    v[15] = kw * (Ax*s1x - Ay*s1y);

    float* g = GS + 16*d;
    #pragma unroll
    for (int k = 0; k < 16; ++k)
        atomAdd(&g[k], INV_SQRT2f * v[k]);
}

// ---------- phase 4: out[d] += INV_SQRT2*PW2 * A(e) . GS[src] ----------
__global__ void k_edge_out(const float* __restrict__ r, const int* __restrict__ esrc,
                           const int* __restrict__ edst, const float* __restrict__ GS,
                           const float* __restrict__ dots, float* __restrict__ out)
{
    int e = blockIdx.x * blockDim.x + threadIdx.x;
    if (e >= EFIX) return;
    int d = edst[e];
    if (d >= NNODE) return;
    int s = esrc[e];
    float s1x = SQRT3f * (r[3*s+0] - r[3*d+0]);
    float s1y = SQRT3f * (r[3*s+1] - r[3*d+1]);
    float s1z = SQRT3f * (r[3*s+2] - r[3*d+2]);

    const float* gp = GS + 16*s;
    float4 g0 = *(const float4*)(gp + 0);
    float4 g1 = *(const float4*)(gp + 4);
    float4 g2 = *(const float4*)(gp + 8);
    float4 g3 = *(const float4*)(gp + 12);

    float d00  = dots[0];  // t1_000 . t2_0e
    float d10  = dots[1];  // t1_110 . t2_0e
    float e011 = dots[2];  // t1_011 . t2_1e
    float e101 = dots[3];  // t1_101 . t2_1e
    float e111 = dots[4];  // t1_111 . t2_1o
    float d22  = dots[5];  // t1_112 . t2_2e

    // q1: (d00*sa + d10*sb) * s1 / sqrt(3)
    float t01 = INV_SQRT3f * (d00 * g0.x + d10 * g0.y);
    float qx = t01 * s1x, qy = t01 * s1y, qz = t01 * s1z;

    // q2: INV_SQRT6 * ( e011 * (p011 x s1) + e101 * (p101 x s1) )
    float p0x = g0.z, p0y = g0.w, p0z = g1.x;
    float p1x = g1.y, p1y = g1.z, p1z = g1.w;
    float c1x = p0y*s1z - p0z*s1y, c1y = p0z*s1x - p0x*s1z, c1z = p0x*s1y - p0y*s1x;
    float c2x = p1y*s1z - p1z*s1y, c2y = p1z*s1x - p1x*s1z, c2z = p1x*s1y - p1y*s1x;
    qx += INV_SQRT6f * (e011*c1x + e101*c2x);
    qy += INV_SQRT6f * (e011*c1y + e101*c2y);
    qz += INV_SQRT6f * (e011*c1z + e101*c2z);

    // q3: INV_SQRT3 * e111 * cr * s0  (s0 == 1)
    qx += INV_SQRT3f * e111 * g2.x;
    qy += INV_SQRT3f * e111 * g2.y;
    qz += INV_SQRT3f * e111 * g2.z;

    // q4: d22 * einsum(w2, s1, C211)
    float w0 = g2.w, w1 = g3.x, w2 = g3.y, w3 = g3.z, w4 = g3.w;
    float B0 = w0*s1y + w3*s1z + (w4 - S3f*w2) * s1x;
    float B1 = w0*s1x + w1*s1z - (S3f*w2 + w4) * s1y;
    float B2 = w1*s1y + 2.0f*S3f*w2*s1z + w3*s1x;
    float kq4 = d22 * CNf;
    qx += kq4 * B0;  qy += kq4 * B1;  qz += kq4 * B2;

    float f = INV_SQRT2f * PW2f;
    atomAdd(&out[3*d+0], f * qx);
    atomAdd(&out[3*d+1], f * qy);
    atomAdd(&out[3*d+2], f * qz);
}

extern "C" void kernel_launch(void* const* d_in, const int* in_sizes, int n_in,
                              void* d_out, int out_size, void* d_ws, size_t ws_size,
                              hipStream_t stream)
{
    const int*   nf     = (const int*)  d_in[0];
    const float* r      = (const float*)d_in[1];
    const int*   esrc   = (const int*)  d_in[2];
    const int*   edst   = (const int*)  d_in[3];
    const float* emb    = (const float*)d_in[4];
    const float* t1_000 = (const float*)d_in[5];
    const float* t1_011 = (const float*)d_in[6];
    const float* t1_101 = (const float*)d_in[7];
    const float* t1_110 = (const float*)d_in[8];
    const float* t1_111 = (const float*)d_in[9];
    const float* t1_112 = (const float*)d_in[10];
    const float* t2_0e  = (const float*)d_in[11];
    const float* t2_1e  = (const float*)d_in[12];
    const float* t2_1o  = (const float*)d_in[13];
    const float* t2_2e  = (const float*)d_in[14];
    float* out = (float*)d_out;
    float* ws  = (float*)d_ws;

    float* dots = ws + DOTS_OFF;
    float* ES   = ws + ES_OFF;
    float* HS   = ws + HS_OFF;
    float* GS   = ws + GS_OFF;

    const int zn = NNODE*24 + NNODE*3;
    k_zero   <<<(zn + 255)/256, 256, 0, stream>>>(ES, out);
    k_dots   <<<1, 32, 0, stream>>>(t1_000, t1_011, t1_101, t1_110, t1_111, t1_112,
                                    t2_0e, t2_1e, t2_1o, t2_2e, dots);
    k_edge_sh <<<(EFIX + 255)/256, 256, 0, stream>>>(r, esrc, edst, ES);
    k_build_hs<<<(NNODE + 255)/256, 256, 0, stream>>>(nf, emb, ES, HS);
    k_edge_mid<<<(EFIX + 255)/256, 256, 0, stream>>>(r, esrc, edst, HS, GS);
    k_edge_out<<<(EFIX + 255)/256, 256, 0, stream>>>(r, esrc, edst, GS, dots, out);
}